// _ProposalLayer_3358664426309
// MI455X (gfx1250) — compile-verified
//
#include <hip/hip_runtime.h>
#include <stdint.h>

#define A_      9
#define H_      384
#define W_      384
#define N_      1327104     /* A_*H_*W_ ; == 162 * 8192 exactly */
#define PRE_    6000
#define POST_   300
#define CAP_    8192
#define CHUNK_  8192
#define NBLK1   (N_ / CHUNK_)   /* 162 */
#define KEY_NEGINF 0x007FFFFFu  /* mapf(-inf) */

typedef uint32_t u32;
typedef unsigned long long u64;

// ---------------------------------------------------------------------------
// order-preserving float -> u32 key map
__device__ __forceinline__ u32 mapf(float f) {
  u32 u = __float_as_uint(f);
  return (u & 0x80000000u) ? ~u : (u | 0x80000000u);
}

// exact replica of reference _clip_boxes expression tree (fp32, no FMA forms)
__device__ __forceinline__ void box_transform(float4 a, float4 d,
                                              float& x1, float& y1,
                                              float& cw, float& ch) {
  float bx = fmaxf(a.x + d.x, 0.0f);
  float by = fmaxf(a.y + d.y, 0.0f);
  float bw = fmaxf(a.z + d.z, 0.0f);
  float bh = fmaxf(a.w + d.w, 0.0f);
  float x2 = (bx + bw) - 1.0f;
  float y2 = (by + bh) - 1.0f;
  x1 = fminf(bx, 383.0f);
  y1 = fminf(by, 383.0f);
  x2 = fminf(x2, 383.0f);
  y2 = fminf(y2, 383.0f);
  cw = (x2 - x1) + 1.0f;
  ch = (y2 - y1) + 1.0f;
}

// ---------------------------------------------------------------------------
// CDNA5 Tensor Data Mover availability (device pass only)
#if defined(__gfx1250__) && __has_builtin(__builtin_amdgcn_tensor_load_to_lds) && __has_builtin(__builtin_amdgcn_s_wait_tensorcnt)
#define HAVE_TDM 1
#else
#define HAVE_TDM 0
#endif

typedef unsigned int uv4 __attribute__((ext_vector_type(4)));
typedef int          iv4 __attribute__((ext_vector_type(4)));
typedef int          iv8 __attribute__((ext_vector_type(8)));

// ---------------------------------------------------------------------------
// Stage NELEM contiguous 32-bit elements global -> LDS via the Tensor Data
// Mover (D#: count=1, type=image, data_size=4B, single-row tile).
// Wave 0 issues the DMA (other waves skip via EXECZ); all waves then wait on
// TENSORcnt and the workgroup barrier. Includes the trailing __syncthreads().
template <int NELEM, int BLK, typename T>
__device__ __forceinline__ void stage_to_lds(const T* __restrict__ g, T* l, int tid) {
  static_assert(sizeof(T) == 4, "32-bit elements only");
  static_assert(NELEM > 0 && NELEM < 65536, "tile_dim0 is 16-bit");
#if HAVE_TDM
  if (tid < 32) {
    u64 ga = (u64)(uintptr_t)g;
    u32 lds_off = (u32)(uintptr_t)(void*)l;
    uv4 g0;
    g0[0] = 1u;                                   // count=1, flags 0, gather off
    g0[1] = lds_off;                              // lds_addr
    g0[2] = (u32)(ga & 0xffffffffu);              // global_addr[31:0]
    g0[3] = (u32)((ga >> 32) & 0x01ffffffu)       // global_addr[56:32]
          | 0x80000000u;                          // type=2 (bits 127:126)
    iv8 g1;
    g1[0] = 0x00020000;                           // wg_mask=0, data_size=4B
    g1[1] = (int)((NELEM & 0xffff) << 16);        // tensor_dim0 lo16
    g1[2] = (int)(((NELEM >> 16) & 0xffff) | (1 << 16)); // dim0 hi16 | tensor_dim1=1
    g1[3] = (int)((NELEM & 0xffff) << 16);        // tile_dim0 = NELEM
    g1[4] = 1;                                    // tile_dim1=1, tile_dim2 unused
    g1[5] = NELEM;                                // tensor_dim0_stride lo32
    g1[6] = 0;
    g1[7] = 0;
    iv4 z4; z4[0] = 0; z4[1] = 0; z4[2] = 0; z4[3] = 0;
    iv8 z8; z8[0] = 0; z8[1] = 0; z8[2] = 0; z8[3] = 0;
    z8[4] = 0; z8[5] = 0; z8[6] = 0; z8[7] = 0;
    __builtin_amdgcn_tensor_load_to_lds(g0, g1, z4, z4, z8, 0);
  }
  __builtin_amdgcn_s_wait_tensorcnt(0);
  __syncthreads();
#else
  for (int i = tid; i < NELEM; i += BLK) l[i] = g[i];
  __syncthreads();
#endif
}

// ---------------------------------------------------------------------------
// K0: zero histogram + select state
__global__ void k_init(u32* __restrict__ hist, u32* __restrict__ state) {
  int t = threadIdx.x;
  if (t < 256) hist[t] = 0u;
  if (t < 16)  state[t] = 0u;
}

// ---------------------------------------------------------------------------
// K1: decode+clip+mask -> key per anchor; MSB-byte histogram.
// Scores tile staged into LDS by the Tensor Data Mover.
__global__ __launch_bounds__(256) void k_keys(const float* __restrict__ scores,
                                              const float4* __restrict__ d4,
                                              const float4* __restrict__ a4,
                                              u32* __restrict__ keys,
                                              u32* __restrict__ ghist) {
  __shared__ float ssc[CHUNK_];
  __shared__ u32 lh[256];
  const int tid = threadIdx.x;
  lh[tid] = 0u;
  const size_t base = (size_t)blockIdx.x * CHUNK_;

  stage_to_lds<CHUNK_, 256>(scores + base, ssc, tid);  // includes barrier

  for (int t = 0; t < CHUNK_ / 256; ++t) {
    int li = tid + t * 256;
    size_t idx = base + (size_t)li;
    float4 d = d4[idx];
    float4 a = a4[idx];
    float x1, y1, cw, ch;
    box_transform(a, d, x1, y1, cw, ch);
    bool kp = (cw >= 2.0f) && (ch >= 2.0f);
    u32 key = kp ? mapf(ssc[li]) : KEY_NEGINF;
    keys[idx] = key;
    atomicAdd(&lh[key >> 24], 1u);
  }
  __syncthreads();
  atomicAdd(&ghist[tid], lh[tid]);
}

// ---------------------------------------------------------------------------
// K2: refine histogram of byte p among keys whose higher bytes match prefix.
// Keys chunk staged into LDS by TDM.
__global__ __launch_bounds__(256) void k_hist(const u32* __restrict__ keys,
                                              u32* __restrict__ hist,
                                              const u32* __restrict__ state,
                                              int p) {
  __shared__ u32 skeys[CHUNK_];
  __shared__ u32 lh[256];
  const int tid = threadIdx.x;
  lh[tid] = 0u;
  const size_t base = (size_t)blockIdx.x * CHUNK_;

  stage_to_lds<CHUNK_, 256>(keys + base, skeys, tid);  // includes barrier

  const u32 prefix = state[0];
  const int shift = 8 * (p + 1);       // 24, 16, or 8
  for (int t = 0; t < CHUNK_ / 256; ++t) {
    u32 k = skeys[tid + t * 256];
    if ((k >> shift) == prefix) atomicAdd(&lh[(k >> (8 * p)) & 255u], 1u);
  }
  __syncthreads();
  atomicAdd(&hist[tid], lh[tid]);
}

// ---------------------------------------------------------------------------
// K3: pick bucket containing the PRE_-th largest key; accumulate prefix
__global__ void k_select(u32* __restrict__ hist, u32* __restrict__ state) {
  if (threadIdx.x != 0 || blockIdx.x != 0) return;
  u32 target = (u32)PRE_ - state[1];
  u32 cum = 0;
  int b = 255;
  for (; b > 0; --b) {
    u32 h = hist[b];
    if (cum + h >= target) break;
    cum += h;
  }
  state[1] += cum;                         // strictly-greater count so far
  state[0] = (state[0] << 8) | (u32)b;     // accumulated prefix (T after 4 passes)
  for (int i = 0; i < 256; ++i) hist[i] = 0u;
}

// ---------------------------------------------------------------------------
// K4: compact all keys >= T as 64-bit (key<<32 | ~idx) items
__global__ __launch_bounds__(256) void k_compact(const u32* __restrict__ keys,
                                                 u32* __restrict__ state,
                                                 u64* __restrict__ items) {
  const u32 T = state[0];
  const int stride = gridDim.x * blockDim.x;
  for (int i = blockIdx.x * blockDim.x + threadIdx.x; i < N_; i += stride) {
    __builtin_prefetch(keys + i + stride, 0, 1);   // global_prefetch on gfx1250
    u32 k = keys[i];
    if (k >= T) {
      u32 pos = atomicAdd(&state[4], 1u);
      if (pos < CAP_)
        items[pos] = (((u64)k) << 32) | (u64)(~(u32)i);
    }
  }
}

// ---------------------------------------------------------------------------
// K5: deterministic LDS bitonic sort (descending) of up to 8192 items;
// emits the exact stable top-6000 (score desc, index asc)
__global__ __launch_bounds__(1024) void k_sort(const u64* __restrict__ items,
                                               const u32* __restrict__ state,
                                               u32* __restrict__ selk,
                                               u32* __restrict__ seli) {
  __shared__ u64 s[CAP_];
  u32 m = state[4];
  if (m > CAP_) m = CAP_;
  for (int i = threadIdx.x; i < CAP_; i += 1024)
    s[i] = (i < (int)m) ? items[i] : 0ull;
  __syncthreads();
  for (int k = 2; k <= CAP_; k <<= 1) {
    for (int j = k >> 1; j > 0; j >>= 1) {
      for (int i = threadIdx.x; i < CAP_; i += 1024) {
        int l = i ^ j;
        if (l > i) {
          u64 a = s[i], b = s[l];
          bool up = ((i & k) == 0);
          if (up ? (a < b) : (a > b)) { s[i] = b; s[l] = a; }
        }
      }
      __syncthreads();
    }
  }
  for (int i = threadIdx.x; i < PRE_; i += 1024) {
    u64 it = s[i];
    selk[i] = (u32)(it >> 32);
    seli[i] = ~((u32)it);
  }
}

// ---------------------------------------------------------------------------
// K6: gather: recompute clipped boxes [x1,y1,w,h] for the 6000 proposals
__global__ __launch_bounds__(256) void k_gather(const u32* __restrict__ seli,
                                                const float4* __restrict__ d4,
                                                const float4* __restrict__ a4,
                                                float* __restrict__ px1,
                                                float* __restrict__ py1,
                                                float* __restrict__ pw,
                                                float* __restrict__ ph) {
  int i = blockIdx.x * blockDim.x + threadIdx.x;
  if (i >= PRE_) return;
  u32 idx = seli[i];
  float x1 = 0.f, y1 = 0.f, cw = 0.f, ch = 0.f;
  if (idx < (u32)N_) {
    float4 d = d4[idx];
    float4 a = a4[idx];
    box_transform(a, d, x1, y1, cw, ch);
  }
  px1[i] = x1; py1[i] = y1; pw[i] = cw; ph[i] = ch;
}

// ---------------------------------------------------------------------------
// K7: sequential NMS (argmax over y2, first-occurrence tie-break) + output.
// The entire 6000x4 proposal block (x1|y1|w|h, contiguous in ws) is staged to
// LDS by one TDM load; the 300-iteration loop then runs entirely out of LDS.
__global__ __launch_bounds__(1024) void k_nms(const float* __restrict__ pbox,
                                              const u32* __restrict__ selk,
                                              float* __restrict__ out) {
  __shared__ float lbox[4 * PRE_];   // [x1 | y1 | w | h] blocks, 96 KB
  __shared__ float ly2[PRE_];
  __shared__ unsigned char lval[PRE_];
  __shared__ u64 wbest[32];
  __shared__ int keepbuf[POST_];
  __shared__ int s_cnt, s_ind, s_stop;
  const int tid = threadIdx.x;

  stage_to_lds<4 * PRE_, 1024>(pbox, lbox, tid);   // includes barrier
  float* lx1 = lbox;
  float* ly1 = lbox + PRE_;
  float* lw  = lbox + 2 * PRE_;
  float* lh  = lbox + 3 * PRE_;

  for (int i = tid; i < PRE_; i += 1024) {
    ly2[i]  = (ly1[i] + lh[i]) - 1.0f;              // y2 = y1 + h - 1, ref order
    lval[i] = (selk[i] > KEY_NEGINF) ? 1 : 0;       // valid = isfinite(score)
  }
  if (tid == 0) s_cnt = 0;
  __syncthreads();

  const u32 NEGKEY = mapf(-1e30f);

  for (int iter = 0; iter < POST_; ++iter) {
    // --- argmax over (valid ? y2 : -1e30), smallest index wins ties ---
    u64 best = ((u64)NEGKEY) << 32;
    for (int i = tid; i < PRE_; i += 1024) {
      float v = lval[i] ? ly2[i] : -1e30f;
      u64 kk = (((u64)mapf(v)) << 32) | (u64)(~(u32)i);
      if (kk > best) best = kk;
    }
    for (int off = 16; off > 0; off >>= 1) {        // wave32 reduction
      u64 o = __shfl_xor(best, off, 32);
      if (o > best) best = o;
    }
    if ((tid & 31) == 0) wbest[tid >> 5] = best;
    __syncthreads();
    if (tid == 0) {
      u64 b = wbest[0];
      for (int w2 = 1; w2 < 32; ++w2) if (wbest[w2] > b) b = wbest[w2];
      u32 hi = (u32)(b >> 32);
      if (hi == NEGKEY) {                           // no valid boxes remain
        s_stop = 1;
      } else {
        int ind = (int)(~(u32)b);
        s_ind = ind;
        keepbuf[s_cnt] = ind;
        s_cnt = s_cnt + 1;
        s_stop = 0;
      }
    }
    __syncthreads();
    if (s_stop) break;

    const int ind = s_ind;
    const float ix1 = lx1[ind];
    const float iy1 = ly1[ind];
    const float ix2 = (ix1 + lw[ind]) - 1.0f;
    const float iy2 = ly2[ind];

    for (int j = tid; j < PRE_; j += 1024) {
      if (!lval[j]) continue;
      float x1 = lx1[j], y1 = ly1[j], w = lw[j], h = lh[j];
      float x2 = (x1 + w) - 1.0f;
      float y2 = ly2[j];
      float xx1 = fmaxf(x1, ix1);
      float yy1 = fmaxf(y1, iy1);
      float xx2 = fminf(x2, ix2);
      float yy2 = fminf(y2, iy2);
      float iw = fmaxf((xx2 - xx1) + 1.0f, 0.0f);
      float ih = fmaxf((yy2 - yy1) + 1.0f, 0.0f);
      float inter = iw * ih;
      float ratio = inter / (w * h);                // inter / areas[j]
      if (!(ratio < 0.7f)) lval[j] = 0;             // NaN/Inf -> suppressed
    }
    if (tid == 0) lval[ind] = 0;
    __syncthreads();
  }

  const int cnt = s_cnt;
  for (int slot = tid; slot < POST_; slot += 1024) {
    float o0 = 0.f, o1 = 0.f, o2 = 0.f, o3 = 0.f;
    if (slot < cnt) {
      int j = keepbuf[slot];
      o0 = lx1[j]; o1 = ly1[j]; o2 = lw[j]; o3 = lh[j];
    }
    out[slot * 4 + 0] = o0;
    out[slot * 4 + 1] = o1;
    out[slot * 4 + 2] = o2;
    out[slot * 4 + 3] = o3;
  }
}

// ---------------------------------------------------------------------------
extern "C" void kernel_launch(void* const* d_in, const int* in_sizes, int n_in,
                              void* d_out, int out_size, void* d_ws, size_t ws_size,
                              hipStream_t stream) {
  const float* scores  = (const float*)d_in[0];
  const float* deltas  = (const float*)d_in[1];
  const float* anchors = (const float*)d_in[2];
  /* d_in[3] = image_metadata (unused by the reference math) */
  float* out = (float*)d_out;

  char* ws = (char*)d_ws;
  size_t off = 0;
  u32* keys  = (u32*)(ws + off); off += (size_t)N_ * 4;     // 5,308,416 B
  u32* hist  = (u32*)(ws + off); off += 256 * 4;
  u32* state = (u32*)(ws + off); off += 64;
  off = (off + 7) & ~(size_t)7;
  u64* items = (u64*)(ws + off); off += (size_t)CAP_ * 8;   // 64 KB
  u32* selk  = (u32*)(ws + off); off += (size_t)PRE_ * 4;
  u32* seli  = (u32*)(ws + off); off += (size_t)PRE_ * 4;
  /* pbox: x1|y1|w|h blocks, MUST stay contiguous (k_nms stages it as one TDM tile) */
  float* pbox = (float*)(ws + off); off += (size_t)4 * PRE_ * 4;
  float* px1 = pbox;
  float* py1 = pbox + PRE_;
  float* pwd = pbox + 2 * PRE_;
  float* phd = pbox + 3 * PRE_;

  const float4* d4 = (const float4*)deltas;
  const float4* a4 = (const float4*)anchors;

  k_init<<<1, 256, 0, stream>>>(hist, state);
  k_keys<<<NBLK1, 256, 0, stream>>>(scores, d4, a4, keys, hist);
  k_select<<<1, 1, 0, stream>>>(hist, state);           // byte 3
  for (int p = 2; p >= 0; --p) {
    k_hist<<<NBLK1, 256, 0, stream>>>(keys, hist, state, p);
    k_select<<<1, 1, 0, stream>>>(hist, state);         // bytes 2,1,0
  }
  k_compact<<<324, 256, 0, stream>>>(keys, state, items);
  k_sort<<<1, 1024, 0, stream>>>(items, state, selk, seli);
  k_gather<<<(PRE_ + 255) / 256, 256, 0, stream>>>(seli, d4, a4, px1, py1, pwd, phd);
  k_nms<<<1, 1024, 0, stream>>>(pbox, selk, out);
}